// Lfass_68745246540208
// MI455X (gfx1250) — compile-verified
//
#include <hip/hip_runtime.h>

// ---------------------------------------------------------------------------
// LFASS loss on MI455X (gfx1250, wave32, WMMA).
// n = 8464 pixels, c = 19 channels. All n x n cosine matrices are streamed
// tile-by-tile through V_WMMA_F32_16X16X4_F32 (K padded 19->20) and reduced
// to per-row / per-column statistic vectors; nothing n x n is materialized.
// ---------------------------------------------------------------------------

#define N_PIX 8464
#define N_TIL 529          // 8464 / 16
#define CPAD  20           // channel pad for K (5 chunks of 4)
#define CQAD  32           // channel pad for ft raw (B operand of w@ft)
#define NCH   19
#define JSPL  8            // column-partition factor per stats kernel
#define INV_N   (1.0f/8464.0f)
#define INV_NM1 (1.0f/8463.0f)

typedef __attribute__((ext_vector_type(2))) float v2f;
typedef __attribute__((ext_vector_type(8))) float v8f;

// sortable-float encoding so unsigned atomicMax orders like float compare
__device__ __forceinline__ unsigned f2s(float f) {
  unsigned b = __float_as_uint(f);
  return (b & 0x80000000u) ? ~b : (b | 0x80000000u);
}
__device__ __forceinline__ float s2f(unsigned s) {
  unsigned b = (s & 0x80000000u) ? (s ^ 0x80000000u) : ~s;
  return __uint_as_float(b);
}

// A fragment for a 16-row strip of X (row-major, stride CPAD), K chunks of 4.
// Lane layout (ISA 7.12.2, 32-bit A 16x4): lanes 0-15 hold K=4kc,4kc+1 of row l,
// lanes 16-31 hold K=4kc+2,4kc+3 of row l.  h2 = 2*(lane>>4).
__device__ __forceinline__ void load_afrag(const float* __restrict__ Xr, int l, int h2, v2f* a) {
#pragma unroll
  for (int kc = 0; kc < 5; ++kc) {
    const float* p = Xr + l * CPAD + 4 * kc + h2;
    a[kc].x = p[0]; a[kc].y = p[1];
  }
}

// 16x16 tile of X @ Y^T via 5 chained f32 WMMAs.  B layout mirrors A
// (both read rows of the row-major operand).  D: lane(l,h) vgpr v ->
// element [row = v + 8h][col = l] of the tile.
__device__ __forceinline__ v8f dtile(const v2f* a, const float* __restrict__ Yr, int l, int h2) {
  v8f c = {0.f,0.f,0.f,0.f,0.f,0.f,0.f,0.f};
#pragma unroll
  for (int kc = 0; kc < 5; ++kc) {
    const float* p = Yr + l * CPAD + 4 * kc + h2;
    v2f b; b.x = p[0]; b.y = p[1];
    c = __builtin_amdgcn_wmma_f32_16x16x4_f32(false, a[kc], false, b, (short)0, c, false, false);
  }
  return c;
}

// ---------------------------------------------------------------------------
// prep: row-normalize fs/ft (F.normalize eps=1e-12), pad to CPAD; raw ft
// padded to CQAD; nearest upsample of labels 730 -> 92.
// ---------------------------------------------------------------------------
__global__ void k_prep(const float* __restrict__ src, const float* __restrict__ trg,
                       const int* __restrict__ labels,
                       float* __restrict__ fsn, float* __restrict__ ftn,
                       float* __restrict__ ftq, int* __restrict__ labn) {
  int i = blockIdx.x * blockDim.x + threadIdx.x;
  if (i >= N_PIX) return;
  float ss = 0.f, st = 0.f;
  for (int ch = 0; ch < NCH; ++ch) {
    float a = src[ch * N_PIX + i]; ss += a * a;
    float b = trg[ch * N_PIX + i]; st += b * b;
  }
  float rs = 1.f / fmaxf(sqrtf(ss), 1e-12f);
  float rt = 1.f / fmaxf(sqrtf(st), 1e-12f);
  for (int ch = 0; ch < NCH; ++ch) {
    float a = src[ch * N_PIX + i];
    float b = trg[ch * N_PIX + i];
    fsn[i * CPAD + ch] = a * rs;
    ftn[i * CPAD + ch] = b * rt;
    ftq[i * CQAD + ch] = b;
  }
  fsn[i * CPAD + NCH] = 0.f;
  ftn[i * CPAD + NCH] = 0.f;
  for (int ch = NCH; ch < CQAD; ++ch) ftq[i * CQAD + ch] = 0.f;
  int r = i / 92, cc = i % 92;
  labn[i] = labels[((r * 730) / 92) * 730 + (cc * 730) / 92];
}

// ---------------------------------------------------------------------------
// Stats pass 1: raw row sums + raw row max/argmax of X @ Y^T.
// One wave per (row tile, column partition); packed-u64 atomicMax gives
// lowest-index tie-break like jnp.argmax.
// ---------------------------------------------------------------------------
__global__ __launch_bounds__(32) void k_stat_summax(
    const float* __restrict__ X, const float* __restrict__ Y,
    float* __restrict__ rowsum, unsigned long long* __restrict__ rowpk) {
  int ti = blockIdx.x, js = blockIdx.y;
  int lane = threadIdx.x & 31, l = lane & 15, h = lane >> 4, h2 = h << 1;
  v2f a[5]; load_afrag(X + ti * 16 * CPAD, l, h2, a);
  float sm[8], mx[8]; int mj[8];
#pragma unroll
  for (int v = 0; v < 8; ++v) { sm[v] = 0.f; mx[v] = -3.4e38f; mj[v] = 0; }
  for (int tj = js; tj < N_TIL; tj += JSPL) {
    v8f c = dtile(a, Y + tj * 16 * CPAD, l, h2);
    int col = tj * 16 + l;
#pragma unroll
    for (int v = 0; v < 8; ++v) {
      float d = c[v];
      sm[v] += d;
      if (d > mx[v] || (d == mx[v] && col < mj[v])) { mx[v] = d; mj[v] = col; }
    }
  }
#pragma unroll
  for (int off = 1; off < 16; off <<= 1) {
#pragma unroll
    for (int v = 0; v < 8; ++v) {
      sm[v] += __shfl_xor(sm[v], off, 32);
      float om = __shfl_xor(mx[v], off, 32);
      int   oj = __shfl_xor(mj[v], off, 32);
      if (om > mx[v] || (om == mx[v] && oj < mj[v])) { mx[v] = om; mj[v] = oj; }
    }
  }
  if (l == 0) {
#pragma unroll
    for (int v = 0; v < 8; ++v) {
      int row = ti * 16 + v + 8 * h;
      atomicAdd(&rowsum[row], sm[v]);
      unsigned long long key =
          ((unsigned long long)f2s(mx[v]) << 32) | (unsigned)(~mj[v]);
      atomicMax(&rowpk[row], key);
    }
  }
}

// Stats pass 2: rowvar_i = sum_j (d_ij - u_j)^2  (u_j = rowsum[j]/n, per COLUMN
// like the torch broadcast).
__global__ __launch_bounds__(32) void k_stat_var(
    const float* __restrict__ X, const float* __restrict__ Y,
    const float* __restrict__ usum, float* __restrict__ varout) {
  int ti = blockIdx.x, js = blockIdx.y;
  int lane = threadIdx.x & 31, l = lane & 15, h = lane >> 4, h2 = h << 1;
  v2f a[5]; load_afrag(X + ti * 16 * CPAD, l, h2, a);
  float vs[8];
#pragma unroll
  for (int v = 0; v < 8; ++v) vs[v] = 0.f;
  for (int tj = js; tj < N_TIL; tj += JSPL) {
    float uu = usum[tj * 16 + l] * INV_N;
    v8f c = dtile(a, Y + tj * 16 * CPAD, l, h2);
#pragma unroll
    for (int v = 0; v < 8; ++v) { float t = c[v] - uu; vs[v] += t * t; }
  }
#pragma unroll
  for (int off = 1; off < 16; off <<= 1)
#pragma unroll
    for (int v = 0; v < 8; ++v) vs[v] += __shfl_xor(vs[v], off, 32);
  if (l == 0)
#pragma unroll
    for (int v = 0; v < 8; ++v)
      atomicAdd(&varout[ti * 16 + v + 8 * h], vs[v]);
}

// Stats pass 3: rowmax of e_ij = (d_ij - u_j) / s_j.
__global__ __launch_bounds__(32) void k_stat_emax(
    const float* __restrict__ X, const float* __restrict__ Y,
    const float* __restrict__ usum, const float* __restrict__ varraw,
    unsigned* __restrict__ emax) {
  int ti = blockIdx.x, js = blockIdx.y;
  int lane = threadIdx.x & 31, l = lane & 15, h = lane >> 4, h2 = h << 1;
  v2f a[5]; load_afrag(X + ti * 16 * CPAD, l, h2, a);
  float mx[8];
#pragma unroll
  for (int v = 0; v < 8; ++v) mx[v] = -3.4e38f;
  for (int tj = js; tj < N_TIL; tj += JSPL) {
    int col = tj * 16 + l;
    float uu  = usum[col] * INV_N;
    float rcp = 1.f / sqrtf(varraw[col] * INV_NM1);
    v8f c = dtile(a, Y + tj * 16 * CPAD, l, h2);
#pragma unroll
    for (int v = 0; v < 8; ++v) { float e = (c[v] - uu) * rcp; mx[v] = fmaxf(mx[v], e); }
  }
#pragma unroll
  for (int off = 1; off < 16; off <<= 1)
#pragma unroll
    for (int v = 0; v < 8; ++v) mx[v] = fmaxf(mx[v], __shfl_xor(mx[v], off, 32));
  if (l == 0)
#pragma unroll
    for (int v = 0; v < 8; ++v)
      atomicMax(&emax[ti * 16 + v + 8 * h], f2s(mx[v]));
}

// Stats pass 4: Z_i = sum_j exp(e_ij - M_i).
__global__ __launch_bounds__(32) void k_stat_z(
    const float* __restrict__ X, const float* __restrict__ Y,
    const float* __restrict__ usum, const float* __restrict__ varraw,
    const unsigned* __restrict__ emax, float* __restrict__ Z) {
  int ti = blockIdx.x, js = blockIdx.y;
  int lane = threadIdx.x & 31, l = lane & 15, h = lane >> 4, h2 = h << 1;
  v2f a[5]; load_afrag(X + ti * 16 * CPAD, l, h2, a);
  float Mr[8], za[8];
#pragma unroll
  for (int v = 0; v < 8; ++v) { Mr[v] = s2f(emax[ti * 16 + v + 8 * h]); za[v] = 0.f; }
  for (int tj = js; tj < N_TIL; tj += JSPL) {
    int col = tj * 16 + l;
    float uu  = usum[col] * INV_N;
    float rcp = 1.f / sqrtf(varraw[col] * INV_NM1);
    v8f c = dtile(a, Y + tj * 16 * CPAD, l, h2);
#pragma unroll
    for (int v = 0; v < 8; ++v) za[v] += __expf((c[v] - uu) * rcp - Mr[v]);
  }
#pragma unroll
  for (int off = 1; off < 16; off <<= 1)
#pragma unroll
    for (int v = 0; v < 8; ++v) za[v] += __shfl_xor(za[v], off, 32);
  if (l == 0)
#pragma unroll
    for (int v = 0; v < 8; ++v)
      atomicAdd(&Z[ti * 16 + v + 8 * h], za[v]);
}

// ---------------------------------------------------------------------------
// Aggregation: num[i][ch] = sum_j exp(e_ij - m_i) * ft[j][ch], den[i] = sum exp.
// The 16x16 exp-weight tile bounces through LDS (stride 17) and re-enters WMMA
// as A against raw ft (B, channels padded to 32 -> two N-tiles).
// ---------------------------------------------------------------------------
__global__ __launch_bounds__(32) void k_agg(
    const float* __restrict__ FTN, const float* __restrict__ FTQ,
    const float* __restrict__ usum, const float* __restrict__ varraw,
    const unsigned* __restrict__ emax,
    float* __restrict__ num, float* __restrict__ den) {
  __shared__ float Wl[16 * 17];
  int ti = blockIdx.x, js = blockIdx.y;
  int lane = threadIdx.x & 31, l = lane & 15, h = lane >> 4, h2 = h << 1;
  v2f a[5]; load_afrag(FTN + ti * 16 * CPAD, l, h2, a);
  float Mr[8], dn[8];
#pragma unroll
  for (int v = 0; v < 8; ++v) { Mr[v] = s2f(emax[ti * 16 + v + 8 * h]); dn[v] = 0.f; }
  v8f acc0 = {0.f,0.f,0.f,0.f,0.f,0.f,0.f,0.f};
  v8f acc1 = {0.f,0.f,0.f,0.f,0.f,0.f,0.f,0.f};
  for (int tj = js; tj < N_TIL; tj += JSPL) {
    int col = tj * 16 + l;
    float uu  = usum[col] * INV_N;
    float rcp = 1.f / sqrtf(varraw[col] * INV_NM1);
    v8f c = dtile(a, FTN + tj * 16 * CPAD, l, h2);
#pragma unroll
    for (int v = 0; v < 8; ++v) {
      float w = __expf((c[v] - uu) * rcp - Mr[v]);
      dn[v] += w;
      Wl[(v + 8 * h) * 17 + l] = w;     // W[striprow][tilecol]
    }
    asm volatile("s_wait_dscnt 0" ::: "memory");  // order LDS store -> load (single wave)
#pragma unroll
    for (int kc = 0; kc < 4; ++kc) {
      int k0 = 4 * kc + h2;
      v2f a2; a2.x = Wl[l * 17 + k0]; a2.y = Wl[l * 17 + k0 + 1];
      const float* B0 = FTQ + (tj * 16 + k0)     * CQAD;
      const float* B1 = FTQ + (tj * 16 + k0 + 1) * CQAD;
      v2f b0; b0.x = B0[l];      b0.y = B1[l];
      acc0 = __builtin_amdgcn_wmma_f32_16x16x4_f32(false, a2, false, b0, (short)0, acc0, false, false);
      v2f b1; b1.x = B0[16 + l]; b1.y = B1[16 + l];
      acc1 = __builtin_amdgcn_wmma_f32_16x16x4_f32(false, a2, false, b1, (short)0, acc1, false, false);
    }
    asm volatile("s_wait_dscnt 0" ::: "memory");  // drain reads before next overwrite
  }
#pragma unroll
  for (int off = 1; off < 16; off <<= 1)
#pragma unroll
    for (int v = 0; v < 8; ++v) dn[v] += __shfl_xor(dn[v], off, 32);
#pragma unroll
  for (int v = 0; v < 8; ++v) {
    int row = ti * 16 + v + 8 * h;
    if (l == 0) atomicAdd(&den[row], dn[v]);
    atomicAdd(&num[row * CQAD + l],      acc0[v]);
    atomicAdd(&num[row * CQAD + 16 + l], acc1[v]);
  }
}

// ftan = normalize(0.5*ft + 0.5*num/den), padded to CPAD.
__global__ void k_finish_agg(const float* __restrict__ trg,
                             const float* __restrict__ num, const float* __restrict__ den,
                             float* __restrict__ ftan) {
  int i = blockIdx.x * blockDim.x + threadIdx.x;
  if (i >= N_PIX) return;
  float d = den[i];
  float v[NCH], s = 0.f;
  for (int ch = 0; ch < NCH; ++ch) {
    float x = 0.5f * trg[ch * N_PIX + i] + 0.5f * (num[i * CQAD + ch] / d);
    v[ch] = x; s += x * x;
  }
  float rn = 1.f / fmaxf(sqrtf(s), 1e-12f);
  for (int ch = 0; ch < NCH; ++ch) ftan[i * CPAD + ch] = v[ch] * rn;
  ftan[i * CPAD + NCH] = 0.f;
}

// Final per-pixel loss terms:  log p1[i,j] = e1[i,j] - M1_i - log Z1_i, etc.
__global__ void k_loss(const unsigned long long* __restrict__ pk1,
                       const unsigned long long* __restrict__ pk2,
                       const float* __restrict__ rowsum1, const float* __restrict__ rowsum2,
                       const float* __restrict__ var1, const float* __restrict__ var2,
                       const unsigned* __restrict__ M1, const unsigned* __restrict__ M2,
                       const float* __restrict__ Z1, const float* __restrict__ Z2,
                       const int* __restrict__ labn, float* __restrict__ accum) {
  int i = blockIdx.x * blockDim.x + threadIdx.x;
  if (i >= N_PIX) return;
  unsigned long long k1 = pk1[i];
  int   j   = (int)(~(unsigned)(k1 & 0xFFFFFFFFu));
  float rm  = s2f((unsigned)(k1 >> 32));
  float e1  = (rm - rowsum1[j] * INV_N) / sqrtf(var1[j] * INV_NM1);
  float lp1 = e1 - s2f(M1[i]) - logf(Z1[i]);
  unsigned long long k2 = pk2[j];
  int   i2  = (int)(~(unsigned)(k2 & 0xFFFFFFFFu));
  float cm  = s2f((unsigned)(k2 >> 32));
  float e2  = (cm - rowsum2[i2] * INV_N) / sqrtf(var2[i2] * INV_NM1);
  float lp2 = e2 - s2f(M2[j]) - logf(Z2[j]);
  if (labn[i] == labn[i2]) {
    atomicAdd(&accum[0], lp1 + lp2);
    atomicAdd(&accum[1], 1.0f);
  }
}

__global__ void k_final(const float* __restrict__ accum, float* __restrict__ out) {
  if (threadIdx.x == 0) out[0] = -accum[0] / accum[1];
}

// ---------------------------------------------------------------------------
extern "C" void kernel_launch(void* const* d_in, const int* in_sizes, int n_in,
                              void* d_out, int out_size, void* d_ws, size_t ws_size,
                              hipStream_t stream) {
  (void)in_sizes; (void)n_in; (void)out_size; (void)ws_size;
  const float* src    = (const float*)d_in[0];
  const float* trg    = (const float*)d_in[1];
  const int*   labels = (const int*)d_in[2];
  float* out = (float*)d_out;
  char* ws = (char*)d_ws;

  size_t o = 0;
  auto alloc = [&](size_t b) { size_t r = o; o = (o + b + 255) & ~(size_t)255; return r; };
  float* fsn  = (float*)(ws + alloc((size_t)N_PIX * CPAD * 4));
  float* ftn  = (float*)(ws + alloc((size_t)N_PIX * CPAD * 4));
  float* ftan = (float*)(ws + alloc((size_t)N_PIX * CPAD * 4));
  float* ftq  = (float*)(ws + alloc((size_t)N_PIX * CQAD * 4));
  float* num  = (float*)(ws + alloc((size_t)N_PIX * CQAD * 4));
  int*   labn = (int*)  (ws + alloc((size_t)N_PIX * 4));
  float* rowsumT = (float*)(ws + alloc((size_t)N_PIX * 4));
  float* varT    = (float*)(ws + alloc((size_t)N_PIX * 4));
  unsigned* mT   = (unsigned*)(ws + alloc((size_t)N_PIX * 4));
  unsigned long long* pkT = (unsigned long long*)(ws + alloc((size_t)N_PIX * 8));
  float* den     = (float*)(ws + alloc((size_t)N_PIX * 4));
  float* rowsum1 = (float*)(ws + alloc((size_t)N_PIX * 4));
  float* rowsum2 = (float*)(ws + alloc((size_t)N_PIX * 4));
  float* var1    = (float*)(ws + alloc((size_t)N_PIX * 4));
  float* var2    = (float*)(ws + alloc((size_t)N_PIX * 4));
  unsigned* M1   = (unsigned*)(ws + alloc((size_t)N_PIX * 4));
  unsigned* M2   = (unsigned*)(ws + alloc((size_t)N_PIX * 4));
  float* Z1      = (float*)(ws + alloc((size_t)N_PIX * 4));
  float* Z2      = (float*)(ws + alloc((size_t)N_PIX * 4));
  unsigned long long* pk1 = (unsigned long long*)(ws + alloc((size_t)N_PIX * 8));
  unsigned long long* pk2 = (unsigned long long*)(ws + alloc((size_t)N_PIX * 8));
  float* accum   = (float*)(ws + alloc(2 * 4));

  hipMemsetAsync(d_ws, 0, o, stream);

  dim3 gv((N_PIX + 255) / 256), bv(256);
  dim3 gs(N_TIL, JSPL), bs(32);

  k_prep<<<gv, bv, 0, stream>>>(src, trg, labels, fsn, ftn, ftq, labn);

  // --- spatial aggregation of ft (d_tt = ftn @ ftn^T) ---
  k_stat_summax<<<gs, bs, 0, stream>>>(ftn, ftn, rowsumT, pkT);
  k_stat_var   <<<gs, bs, 0, stream>>>(ftn, ftn, rowsumT, varT);
  k_stat_emax  <<<gs, bs, 0, stream>>>(ftn, ftn, rowsumT, varT, mT);
  k_agg        <<<gs, bs, 0, stream>>>(ftn, ftq, rowsumT, varT, mT, num, den);
  k_finish_agg <<<gv, bv, 0, stream>>>(trg, num, den, ftan);

  // --- d1 = fsn @ ftan^T ; d2 = d1^T via swapped operands ---
  k_stat_summax<<<gs, bs, 0, stream>>>(fsn,  ftan, rowsum1, pk1);
  k_stat_summax<<<gs, bs, 0, stream>>>(ftan, fsn,  rowsum2, pk2);
  k_stat_var   <<<gs, bs, 0, stream>>>(fsn,  ftan, rowsum1, var1);
  k_stat_var   <<<gs, bs, 0, stream>>>(ftan, fsn,  rowsum2, var2);
  k_stat_emax  <<<gs, bs, 0, stream>>>(fsn,  ftan, rowsum1, var1, M1);
  k_stat_emax  <<<gs, bs, 0, stream>>>(ftan, fsn,  rowsum2, var2, M2);
  k_stat_z     <<<gs, bs, 0, stream>>>(fsn,  ftan, rowsum1, var1, M1, Z1);
  k_stat_z     <<<gs, bs, 0, stream>>>(ftan, fsn,  rowsum2, var2, M2, Z2);

  k_loss <<<gv, bv, 0, stream>>>(pk1, pk2, rowsum1, rowsum2, var1, var2,
                                 M1, M2, Z1, Z2, labn, accum);
  k_final<<<1, 32, 0, stream>>>(accum, out);
}